// PointNetSetAbstraction_44152263803316
// MI455X (gfx1250) — compile-verified
//
#include <hip/hip_runtime.h>
#include <hip/hip_bf16.h>

typedef float v2f __attribute__((ext_vector_type(2)));
typedef float v8f __attribute__((ext_vector_type(8)));

#define B_     8
#define N_     8192
#define C_     32
#define S_     2048
#define K_     32
#define ROWS   (B_ * S_ * K_)      // 524288
#define NEWXYZ_ELEMS (B_ * S_ * 3) // 49152
#define MMBLOCKS (ROWS / 128)      // 4096 blocks, 8 waves x 16 rows each

// ---------------------------------------------------------------------------
// 1) Farthest point sampling: one workgroup per batch, xyz staged in LDS.
//    Writes new_xyz directly into d_out[0 .. 49151].
// ---------------------------------------------------------------------------
__global__ __launch_bounds__(1024) void fps_kernel(const float* __restrict__ xyz,
                                                   float* __restrict__ newxyz) {
  extern __shared__ float sm[];
  float* px = sm;
  float* py = sm + N_;
  float* pz = sm + 2 * N_;
  __shared__ float redv[32];
  __shared__ int   redi[32];
  __shared__ int   bcast;

  const int b = blockIdx.x;
  const int t = threadIdx.x;
  const float* base = xyz + (size_t)b * N_ * 3;
  for (int i = t; i < N_; i += 1024) {
    px[i] = base[3 * i + 0];
    py[i] = base[3 * i + 1];
    pz[i] = base[3 * i + 2];
  }
  float dist[8];
#pragma unroll
  for (int j = 0; j < 8; ++j) dist[j] = 1e10f;
  __syncthreads();

  int f = 0;
  const int lane = t & 31;
  const int wv = t >> 5;
  for (int s = 0; s < S_; ++s) {
    const float cx = px[f], cy = py[f], cz = pz[f];
    if (t == 0) {
      newxyz[(b * S_ + s) * 3 + 0] = cx;
      newxyz[(b * S_ + s) * 3 + 1] = cy;
      newxyz[(b * S_ + s) * 3 + 2] = cz;
    }
    float best = -1.0f;
    int bi = 0;
#pragma unroll
    for (int j = 0; j < 8; ++j) {
      const int i = t + j * 1024;
      const float dx = px[i] - cx, dy = py[i] - cy, dz = pz[i] - cz;
      const float d = dx * dx + dy * dy + dz * dz;
      const float nd = fminf(dist[j], d);
      dist[j] = nd;
      if (nd > best || (nd == best && i < bi)) { best = nd; bi = i; }
    }
    // wave32 argmax (tie -> lower index, matching argmax-first semantics)
#pragma unroll
    for (int off = 16; off > 0; off >>= 1) {
      const float v2 = __shfl_xor(best, off, 32);
      const int   i2 = __shfl_xor(bi, off, 32);
      if (v2 > best || (v2 == best && i2 < bi)) { best = v2; bi = i2; }
    }
    if (lane == 0) { redv[wv] = best; redi[wv] = bi; }
    __syncthreads();
    if (wv == 0) {
      best = redv[lane];
      bi = redi[lane];
#pragma unroll
      for (int off = 16; off > 0; off >>= 1) {
        const float v2 = __shfl_xor(best, off, 32);
        const int   i2 = __shfl_xor(bi, off, 32);
        if (v2 > best || (v2 == best && i2 < bi)) { best = v2; bi = i2; }
      }
      if (lane == 0) bcast = bi;
    }
    __syncthreads();
    f = bcast;
    __syncthreads();
  }
}

// ---------------------------------------------------------------------------
// 2) Ball query: one wave per center; ordered first-K selection via shfl scan.
// ---------------------------------------------------------------------------
__global__ __launch_bounds__(128) void ballquery_kernel(const float* __restrict__ xyz,
                                                        const float* __restrict__ newxyz,
                                                        int* __restrict__ gidx) {
  __shared__ int ibuf[4][K_];
  const int lane = threadIdx.x & 31;
  const int w = threadIdx.x >> 5;
  const int c = blockIdx.x * 4 + w;  // 0..16383
  const int b = c >> 11;
  const float cx = newxyz[c * 3 + 0];
  const float cy = newxyz[c * 3 + 1];
  const float cz = newxyz[c * 3 + 2];
  const float* base = xyz + (size_t)b * N_ * 3;
  const float r2 = 0.4f * 0.4f;

  int cnt = 0;
  float bestd = 3.4e38f;
  int besti = 0;
  for (int chunk = 0; chunk < N_ / 32 && cnt < K_; ++chunk) {
    const int i = chunk * 32 + lane;
    const float dx = base[3 * i + 0] - cx;
    const float dy = base[3 * i + 1] - cy;
    const float dz = base[3 * i + 2] - cz;
    const float d = dx * dx + dy * dy + dz * dz;
    const bool in = d < r2;
    // inclusive scan of predicate across the wave
    int incl = in ? 1 : 0;
#pragma unroll
    for (int off = 1; off < 32; off <<= 1) {
      const int v = __shfl_up(incl, off, 32);
      if (lane >= off) incl += v;
    }
    const int total = __shfl(incl, 31, 32);
    if (in) {
      const int pos = cnt + incl - 1;
      if (pos < K_) ibuf[w][pos] = i;
    }
    cnt += total;
    if (d < bestd) { bestd = d; besti = i; }
  }
  // argmin fallback reduction (only meaningful when cnt==0, full scan done)
#pragma unroll
  for (int off = 16; off > 0; off >>= 1) {
    const float v2 = __shfl_xor(bestd, off, 32);
    const int   i2 = __shfl_xor(besti, off, 32);
    if (v2 < bestd || (v2 == bestd && i2 < besti)) { bestd = v2; besti = i2; }
  }
  __syncthreads();
  const int first = (cnt == 0) ? besti : ibuf[w][0];
  const int v = (lane < cnt) ? ibuf[w][lane] : first;
  gidx[c * K_ + lane] = v;
}

// ---------------------------------------------------------------------------
// 3) Build X0: gather grouped (xyz - center, features) into [ROWS][36] (pad=0)
// ---------------------------------------------------------------------------
__global__ __launch_bounds__(256) void buildx0_kernel(const float* __restrict__ xyz,
                                                      const float* __restrict__ feat,
                                                      const float* __restrict__ newxyz,
                                                      const int* __restrict__ gidx,
                                                      float* __restrict__ X0) {
  const int e = blockIdx.x * 256 + threadIdx.x;  // ROWS*36 = 18,874,368 exactly
  const int col = e % 36;
  const int row = e / 36;
  const int b = row >> 16;          // / (S_*K_)
  const int rem = row & 65535;
  const int s = rem >> 5;
  const int g = gidx[row];
  float v;
  if (col < 3) {
    v = xyz[((size_t)b * N_ + g) * 3 + col] - newxyz[(b * S_ + s) * 3 + col];
  } else if (col < 35) {
    v = feat[((size_t)b * N_ + g) * C_ + (col - 3)];
  } else {
    v = 0.0f;
  }
  X0[(size_t)row * 36 + col] = v;
}

// ---------------------------------------------------------------------------
// 4) Fused matmul (fp32 WMMA 16x16x4) + bias + deterministic BN stats.
//    Weights staged TRANSPOSED + padded in LDS: wlds[n][k], row stride
//    CINP+4 dwords -> lane n hits bank 4n (conflict-free) and the two K
//    values per WMMA fragment are one aligned ds_load_b64.
//    Each wave: 16 rows x COUT. Block: 8 waves = 128 rows.
// ---------------------------------------------------------------------------
template <int CINP, int CINR, int COUT, bool INBN>
__global__ __launch_bounds__(256) void mlp_matmul_kernel(
    const float* __restrict__ X, const float* __restrict__ W,
    const float* __restrict__ bias,
    const float* __restrict__ insc, const float* __restrict__ insh,
    float* __restrict__ H, float* __restrict__ partial /* [blocks][2][COUT] */) {
  constexpr int WST = CINP + 4;        // padded LDS row stride (dwords)
  __shared__ float wlds[COUT * WST];   // W (row-major [n][k]), zero-padded on K
  __shared__ float wsum[8][COUT];
  __shared__ float wsq[8][COUT];
  __shared__ float scl[CINP];
  __shared__ float shf[CINP];

  const int tid = threadIdx.x;
  for (int i = tid; i < COUT * CINP; i += 256) {
    const int n = i / CINP, k = i % CINP;
    wlds[n * WST + k] = (k < CINR) ? W[n * CINR + k] : 0.0f;
  }
  if (INBN) {
    for (int i = tid; i < CINP; i += 256) {
      scl[i] = (i < CINR) ? insc[i] : 0.0f;
      shf[i] = (i < CINR) ? insh[i] : 0.0f;
    }
  }
  __syncthreads();

  const int wave = tid >> 5;
  const int lane = tid & 31;
  const int m = lane & 15;
  const int kh = (lane >> 4) << 1;       // 0 or 2: K-pair held by this lane half
  const int mrow = (lane >> 4) << 3;     // C/D layout: +8 rows for upper half
  const int rowbase = (blockIdx.x * 8 + wave) * 16;
  const float* xr = X + (size_t)(rowbase + m) * CINP;

  constexpr int CT = COUT / 16;
  v8f acc[CT] = {};

#pragma unroll
  for (int k0 = 0; k0 < CINP; k0 += 4) {
    v2f A = *(const v2f*)&xr[k0 + kh];          // 8B-aligned global b64 load
    if (INBN) {
      A.x = fmaxf(0.0f, fmaf(A.x, scl[k0 + kh], shf[k0 + kh]));
      A.y = fmaxf(0.0f, fmaf(A.y, scl[k0 + kh + 1], shf[k0 + kh + 1]));
    }
#pragma unroll
    for (int t2 = 0; t2 < CT; ++t2) {
      const v2f Bf = *(const v2f*)&wlds[(t2 * 16 + m) * WST + k0 + kh];  // ds_load_b64
      acc[t2] = __builtin_amdgcn_wmma_f32_16x16x4_f32(
          false, A, false, Bf, (short)0, acc[t2], false, false);
    }
  }

  // bias + store + per-wave column sums (deterministic reduction path)
#pragma unroll
  for (int t2 = 0; t2 < CT; ++t2) {
    const int n = t2 * 16 + m;
    const float bv = bias[n];
    float s1 = 0.0f, s2 = 0.0f;
#pragma unroll
    for (int r = 0; r < 8; ++r) {
      const float h = acc[t2][r] + bv;
      H[(size_t)(rowbase + r + mrow) * COUT + n] = h;
      s1 += h;
      s2 += h * h;
    }
    s1 += __shfl_xor(s1, 16, 32);  // combine the two lane halves (same column)
    s2 += __shfl_xor(s2, 16, 32);
    if (lane < 16) { wsum[wave][n] = s1; wsq[wave][n] = s2; }
  }
  __syncthreads();
  for (int i = tid; i < COUT; i += 256) {
    float a = 0.0f, q = 0.0f;
#pragma unroll
    for (int w2 = 0; w2 < 8; ++w2) { a += wsum[w2][i]; q += wsq[w2][i]; }
    partial[(size_t)blockIdx.x * 2 * COUT + i] = a;
    partial[(size_t)blockIdx.x * 2 * COUT + COUT + i] = q;
  }
}

// ---------------------------------------------------------------------------
// 5) Reduce per-block stats in fixed order, produce BN scale/shift.
// ---------------------------------------------------------------------------
__global__ void bn_reduce_finalize_kernel(const float* __restrict__ partial,
                                          const float* __restrict__ g,
                                          const float* __restrict__ be,
                                          float* __restrict__ sc,
                                          float* __restrict__ sh,
                                          int cout, int nblocks, float inv_n) {
  const int c = threadIdx.x;
  if (c >= cout) return;
  float s1 = 0.0f, s2 = 0.0f;
  for (int b = 0; b < nblocks; ++b) {
    s1 += partial[(size_t)b * 2 * cout + c];
    s2 += partial[(size_t)b * 2 * cout + cout + c];
  }
  const float mean = s1 * inv_n;
  const float var = s2 * inv_n - mean * mean;
  const float rstd = rsqrtf(var + 1e-5f);
  const float s = g[c] * rstd;
  sc[c] = s;
  sh[c] = be[c] - mean * s;
}

// ---------------------------------------------------------------------------
// 6) Apply final BN+ReLU and max-pool over K neighbors.
// ---------------------------------------------------------------------------
__global__ __launch_bounds__(128) void maxpool_kernel(const float* __restrict__ h2,
                                                      const float* __restrict__ sc,
                                                      const float* __restrict__ sh,
                                                      float* __restrict__ out) {
  const int c = threadIdx.x;          // 0..127
  const int p = blockIdx.x;           // b*S_+s, 0..16383
  const size_t row0 = (size_t)p * K_;
  const float s = sc[c], t = sh[c];
  float m = -3.4e38f;
#pragma unroll
  for (int k = 0; k < K_; ++k) {
    const float v = fmaxf(0.0f, fmaf(h2[(row0 + k) * 128 + c], s, t));
    m = fmaxf(m, v);
  }
  out[NEWXYZ_ELEMS + (size_t)p * 128 + c] = m;
}

// ---------------------------------------------------------------------------
extern "C" void kernel_launch(void* const* d_in, const int* in_sizes, int n_in,
                              void* d_out, int out_size, void* d_ws, size_t ws_size,
                              hipStream_t stream) {
  (void)in_sizes; (void)n_in; (void)out_size; (void)ws_size;
  const float* xyz  = (const float*)d_in[0];
  const float* feat = (const float*)d_in[1];
  const float* w0 = (const float*)d_in[2];
  const float* b0 = (const float*)d_in[3];
  const float* g0 = (const float*)d_in[4];
  const float* e0 = (const float*)d_in[5];
  const float* w1 = (const float*)d_in[6];
  const float* b1 = (const float*)d_in[7];
  const float* g1 = (const float*)d_in[8];
  const float* e1 = (const float*)d_in[9];
  const float* w2 = (const float*)d_in[10];
  const float* b2 = (const float*)d_in[11];
  const float* g2 = (const float*)d_in[12];
  const float* e2 = (const float*)d_in[13];
  float* out = (float*)d_out;

  auto up = [](size_t x) { return (x + 255) & ~(size_t)255; };
  char* p = (char*)d_ws;
  int* gidx = (int*)p;        p += up((size_t)ROWS * sizeof(int));
  float* partial = (float*)p; p += up((size_t)MMBLOCKS * 2 * 128 * sizeof(float));
  float* sc0 = (float*)p;     p += up(64 * sizeof(float));
  float* sh0 = (float*)p;     p += up(64 * sizeof(float));
  float* sc1 = (float*)p;     p += up(64 * sizeof(float));
  float* sh1 = (float*)p;     p += up(64 * sizeof(float));
  float* sc2 = (float*)p;     p += up(128 * sizeof(float));
  float* sh2 = (float*)p;     p += up(128 * sizeof(float));
  float* X0 = (float*)p;      p += up((size_t)ROWS * 36 * sizeof(float));
  float* h0 = (float*)p;      p += up((size_t)ROWS * 64 * sizeof(float));
  float* h1 = (float*)p;      p += up((size_t)ROWS * 64 * sizeof(float));
  float* h2 = (float*)p;      p += up((size_t)ROWS * 128 * sizeof(float));

  const float inv_n = 1.0f / (float)ROWS;

  fps_kernel<<<B_, 1024, 3 * N_ * sizeof(float), stream>>>(xyz, out);
  ballquery_kernel<<<(B_ * S_) / 4, 128, 0, stream>>>(xyz, out, gidx);
  buildx0_kernel<<<(ROWS * 36) / 256, 256, 0, stream>>>(xyz, feat, out, gidx, X0);

  mlp_matmul_kernel<36, 35, 64, false><<<MMBLOCKS, 256, 0, stream>>>(
      X0, w0, b0, nullptr, nullptr, h0, partial);
  bn_reduce_finalize_kernel<<<1, 64, 0, stream>>>(partial, g0, e0, sc0, sh0, 64, MMBLOCKS, inv_n);

  mlp_matmul_kernel<64, 64, 64, true><<<MMBLOCKS, 256, 0, stream>>>(
      h0, w1, b1, sc0, sh0, h1, partial);
  bn_reduce_finalize_kernel<<<1, 64, 0, stream>>>(partial, g1, e1, sc1, sh1, 64, MMBLOCKS, inv_n);

  mlp_matmul_kernel<64, 64, 128, true><<<MMBLOCKS, 256, 0, stream>>>(
      h1, w2, b2, sc1, sh1, h2, partial);
  bn_reduce_finalize_kernel<<<1, 128, 0, stream>>>(partial, g2, e2, sc2, sh2, 128, MMBLOCKS, inv_n);

  maxpool_kernel<<<B_ * S_, 128, 0, stream>>>(h2, sc2, sh2, out);
}